// GraphCondSelective_652835029231
// MI455X (gfx1250) — compile-verified
//
#include <hip/hip_runtime.h>
#include <hip/hip_bf16.h>
#include <math.h>

// ---------------- problem constants (match reference) ----------------
#define GN      48000          // nodes
#define GB      16             // graphs
#define NPG     3000           // nodes per graph
#define GE      800000         // edges
#define GD      128            // feature dim (D_IN == D_H == D_LAT)
#define GK      16             // base nodes per graph
#define DCOND   32
#define NC      2
#define ZW      (GD * (1 + GK))   // 2176, lin1 input width

typedef __attribute__((ext_vector_type(2))) float v2f;
typedef __attribute__((ext_vector_type(8))) float v8f;

// async-to-LDS builtin pointer types (match builtin prototype:
// pointer to 16B int vector; src in global AS(1), dst in LDS AS(3))
typedef int v4i __attribute__((vector_size(16)));
typedef __attribute__((address_space(1))) v4i* g4ptr;
typedef __attribute__((address_space(3))) v4i* l4ptr;

// LDS padding strides (floats) to dodge 64-bank conflicts
#define WPAD 132    // transposed W tile stride: sWt[c*WPAD + k]
#define APAD 132    // A tile stride:            sA[r*APAD + k]

// ---------------------------------------------------------------------
__global__ void zero_f32(float* __restrict__ p, int n) {
    int t = blockIdx.x * blockDim.x + threadIdx.x;
    if (t < n) p[t] = 0.0f;
}

// degree count: atomic +1 per edge destination
__global__ void deg_count(const int* __restrict__ ei, float* __restrict__ deg, int ne) {
    int e = blockIdx.x * blockDim.x + threadIdx.x;
    if (e < ne) atomicAdd(&deg[ei[ne + e]], 1.0f);
}

// dinv[n] = rsqrt(deg[n] + 1)   (in-place over deg buffer)
__global__ void deg_rsqrt(float* __restrict__ d, int n) {
    int t = blockIdx.x * blockDim.x + threadIdx.x;
    if (t < n) d[t] = rsqrtf(d[t] + 1.0f);
}

// ---------------------------------------------------------------------
// hlin = (doRelu ? relu(hin) : hin) @ W      [GN,128] x [128,128]
// 256 threads = 8 waves; workgroup computes 32 rows x 128 cols per tile.
// W staged transposed in LDS once; A tiles double-buffered via
// GLOBAL_LOAD_ASYNC_TO_LDS_B128 (ASYNCcnt), prefetching tile t+1 during
// the 64 WMMAs of tile t. ReLU applied on A fragments (2 v_max per step).
__global__ void gcn_gemm_wmma(const float* __restrict__ hin,
                              const float* __restrict__ W,
                              float* __restrict__ hlin,
                              int nRowBlocks, int doRelu) {
    extern __shared__ float smem[];
    float* sWt = smem;                       // [128][WPAD] transposed
    float* sA0 = smem + 128 * WPAD;          // [32][APAD] tile buffer 0
    float* sA1 = sA0 + 32 * APAD;            // [32][APAD] tile buffer 1

    const int tid  = threadIdx.x;
    const int lane = tid & 31;
    const int wid  = tid >> 5;
    const int hi   = lane >> 4;              // K-pair select (ISA f32 A layout)
    const int m    = lane & 15;

    // stage W (row-major in memory) transposed into LDS (synchronous, once)
    for (int i = tid; i < 128 * 32; i += 256) {
        const int k = i >> 5;
        const int c = (i & 31) << 2;
        const float4 w4 = *(const float4*)(W + k * GD + c);
        sWt[(c + 0) * WPAD + k] = w4.x;
        sWt[(c + 1) * WPAD + k] = w4.y;
        sWt[(c + 2) * WPAD + k] = w4.z;
        sWt[(c + 3) * WPAD + k] = w4.w;
    }

    // this thread's 4 float4 slots within a 32x128 tile
    const int sc = (tid & 31) << 2;

    // async-stage one 32x128 A tile into buf (4 x b128 per thread)
    auto issue_tile = [&](int r0, float* buf) {
#pragma unroll
        for (int j = 0; j < 4; ++j) {
            const int r = (tid + j * 256) >> 5;     // 0..31
            __builtin_amdgcn_global_load_async_to_lds_b128(
                (g4ptr)(hin + (size_t)(r0 + r) * GD + sc),
                (l4ptr)(buf + r * APAD + sc), 0, 0);
        }
    };

    const int rt = wid >> 2;                 // 0..1  row tile within block
    const int c0 = (wid & 3) * 32;           // 0,32,64,96

    const int rb0 = blockIdx.x;
    if (rb0 < nRowBlocks) issue_tile(rb0 * 32, sA0);

    int parity = 0;
    for (int rb = rb0; rb < nRowBlocks; rb += gridDim.x) {
        const int r0 = rb * 32;
        float* cur = parity ? sA1 : sA0;
        float* nxt = parity ? sA0 : sA1;

        __syncthreads();   // retire readers of `nxt` (and W staging on iter 0)
        const int rbn = rb + gridDim.x;
        if (rbn < nRowBlocks) {
            issue_tile(rbn * 32, nxt);
            // 8 outstanding max; first 4 (current tile) complete in order
            asm volatile("s_wait_asynccnt 0x4" ::: "memory");
        } else {
            asm volatile("s_wait_asynccnt 0x0" ::: "memory");
        }
        __syncthreads();   // all threads' current-tile data visible in LDS

        v8f acc0 = {};
        v8f acc1 = {};
        const float* aRow  = cur + (rt * 16 + m) * APAD;
        const float* bCol0 = sWt + (c0 + m) * WPAD;
        const float* bCol1 = sWt + (c0 + 16 + m) * WPAD;
        if (doRelu) {
#pragma unroll 8
            for (int k = 0; k < GD; k += 4) {
                const int kk = k + 2 * hi;
                v2f a, b0, b1;
                a.x  = fmaxf(aRow[kk], 0.0f);
                a.y  = fmaxf(aRow[kk + 1], 0.0f);
                b0.x = bCol0[kk];     b0.y = bCol0[kk + 1];
                b1.x = bCol1[kk];     b1.y = bCol1[kk + 1];
                acc0 = __builtin_amdgcn_wmma_f32_16x16x4_f32(false, a, false, b0,
                                                             (short)0, acc0, false, false);
                acc1 = __builtin_amdgcn_wmma_f32_16x16x4_f32(false, a, false, b1,
                                                             (short)0, acc1, false, false);
            }
        } else {
#pragma unroll 8
            for (int k = 0; k < GD; k += 4) {
                const int kk = k + 2 * hi;
                v2f a, b0, b1;
                a.x  = aRow[kk];      a.y  = aRow[kk + 1];
                b0.x = bCol0[kk];     b0.y = bCol0[kk + 1];
                b1.x = bCol1[kk];     b1.y = bCol1[kk + 1];
                acc0 = __builtin_amdgcn_wmma_f32_16x16x4_f32(false, a, false, b0,
                                                             (short)0, acc0, false, false);
                acc1 = __builtin_amdgcn_wmma_f32_16x16x4_f32(false, a, false, b1,
                                                             (short)0, acc1, false, false);
            }
        }
        // C/D layout: VGPR i -> row (i + hi*8), lanes 0..15 -> col c0+m
        const int rowBase = r0 + rt * 16 + hi * 8;
#pragma unroll
        for (int i = 0; i < 8; ++i) {
            hlin[(size_t)(rowBase + i) * GD + c0 + m]      = acc0[i];
            hlin[(size_t)(rowBase + i) * GD + c0 + 16 + m] = acc1[i];
        }
        parity ^= 1;
    }
}

// agg[n][c] = hlin[n][c] * dinv[n]^2 + b[c]   (self-loop term + bias)
__global__ void gcn_init_agg(const float* __restrict__ hlin,
                             const float* __restrict__ dinv,
                             const float* __restrict__ b,
                             float* __restrict__ agg, int n4) {
    int t = blockIdx.x * blockDim.x + threadIdx.x;   // one per float4
    if (t >= n4) return;
    const int n  = t >> 5;
    const int c4 = (t & 31) << 2;
    const float di = dinv[n];
    const float s  = di * di;
    const float4 v  = *(const float4*)(hlin + (size_t)n * GD + c4);
    const float4 bb = *(const float4*)(b + c4);
    float4 o;
    o.x = v.x * s + bb.x; o.y = v.y * s + bb.y;
    o.z = v.z * s + bb.z; o.w = v.w * s + bb.w;
    *(float4*)(agg + (size_t)n * GD + c4) = o;
}

// one 32-lane group per edge; lane owns 4 channels; L2-side f32 atomics
__global__ void gcn_edge_scatter(const float* __restrict__ hlin,
                                 const float* __restrict__ dinv,
                                 const int* __restrict__ ei,
                                 float* __restrict__ agg, int ne) {
    const int t    = blockIdx.x * blockDim.x + threadIdx.x;
    const int e    = t >> 5;
    const int lane = t & 31;
    if (e >= ne) return;
    const int s = ei[e];
    const int d = ei[ne + e];
    const float nrm = dinv[s] * dinv[d];
    const float4 v = *(const float4*)(hlin + (size_t)s * GD + lane * 4);
    float* out = agg + (size_t)d * GD + lane * 4;
    atomicAdd(out + 0, v.x * nrm);
    atomicAdd(out + 1, v.y * nrm);
    atomicAdd(out + 2, v.z * nrm);
    atomicAdd(out + 3, v.w * nrm);
}

// mean pool into z cols [2048,2176); grid (B, 24), 125 nodes per block
__global__ void pool_mean(const float* __restrict__ h, float* __restrict__ z) {
    const int g  = blockIdx.x;
    const int ck = blockIdx.y;
    const int c  = threadIdx.x;             // 0..127
    const int n0 = g * NPG + ck * 125;
    float s = 0.0f;
    for (int i = 0; i < 125; ++i) s += h[(size_t)(n0 + i) * GD + c];
    atomicAdd(&z[g * ZW + GK * GD + c], s * (1.0f / (float)NPG));
}

// masked gather of base nodes into z cols [0,2048); grid B*K, 128 threads
__global__ void gather_base(const float* __restrict__ h, const int* __restrict__ base,
                            float* __restrict__ z) {
    const int g = blockIdx.x >> 4;
    const int k = blockIdx.x & 15;
    const int c = threadIdx.x;
    const int bn = base[g * GK + k];
    float v = 0.0f;
    if (bn > 0) v = h[(size_t)(g * NPG + bn) * GD + c];
    z[g * ZW + k * GD + c] = v;
}

// z2 = relu(z @ lin1_W + lin1_b)   [16,2176] x [2176,128]; single WG, 8 waves
__global__ void lin1_wmma(const float* __restrict__ z, const float* __restrict__ Wl,
                          const float* __restrict__ bl, float* __restrict__ z2) {
    const int tid  = threadIdx.x;
    const int lane = tid & 31;
    const int wid  = tid >> 5;
    const int hi   = lane >> 4;
    const int m    = lane & 15;
    const int c0   = wid * 16;
    v8f acc = {};
    for (int k = 0; k < ZW; k += 4) {
        const int kk = k + 2 * hi;
        v2f a, b;
        a.x = z[m * ZW + kk];
        a.y = z[m * ZW + kk + 1];
        b.x = Wl[(size_t)kk * GD + c0 + m];
        b.y = Wl[(size_t)(kk + 1) * GD + c0 + m];
        acc = __builtin_amdgcn_wmma_f32_16x16x4_f32(false, a, false, b,
                                                    (short)0, acc, false, false);
    }
    const float bias = bl[c0 + m];
#pragma unroll
    for (int i = 0; i < 8; ++i) {
        const int row = i + hi * 8;
        z2[row * GD + c0 + m] = fmaxf(acc[i] + bias, 0.0f);
    }
}

// out = [z2, cond] @ lin2_W + lin2_b   (tiny: 32 useful threads)
__global__ void lin2_small(const float* __restrict__ z2, const float* __restrict__ cond,
                           const float* __restrict__ W2, const float* __restrict__ b2,
                           float* __restrict__ out) {
    const int t = threadIdx.x;
    if (t >= GB * NC) return;
    const int g = t >> 1;
    const int j = t & 1;
    float s = b2[j];
    for (int c = 0; c < GD; ++c)    s += z2[g * GD + c]   * W2[c * NC + j];
    for (int c = 0; c < DCOND; ++c) s += cond[g * DCOND + c] * W2[(GD + c) * NC + j];
    out[g * NC + j] = s;
}

// ---------------------------------------------------------------------
extern "C" void kernel_launch(void* const* d_in, const int* in_sizes, int n_in,
                              void* d_out, int out_size, void* d_ws, size_t ws_size,
                              hipStream_t stream) {
    (void)in_sizes; (void)n_in; (void)out_size; (void)ws_size;

    const float* x    = (const float*)d_in[0];
    const int*   ei   = (const int*)  d_in[1];   // [2,E] flat
    const int*   base = (const int*)  d_in[3];
    const float* cond = (const float*)d_in[4];
    const float* Ws[5] = { (const float*)d_in[5],  (const float*)d_in[7],
                           (const float*)d_in[9],  (const float*)d_in[11],
                           (const float*)d_in[13] };
    const float* bs[5] = { (const float*)d_in[6],  (const float*)d_in[8],
                           (const float*)d_in[10], (const float*)d_in[12],
                           (const float*)d_in[14] };
    const float* lin1W = (const float*)d_in[15];
    const float* lin1b = (const float*)d_in[16];
    const float* lin2W = (const float*)d_in[17];
    const float* lin2b = (const float*)d_in[18];
    float* out = (float*)d_out;

    // workspace layout (floats), 256B-aligned chunks
    float* ws = (float*)d_ws;
    size_t off = 0;
    auto take = [&](size_t nFloats) {
        float* p = ws + off;
        off += (nFloats + 63) & ~(size_t)63;
        return p;
    };
    float* dinv = take(GN);                   // degree, then rsqrt in place
    float* bufA = take((size_t)GN * GD);      // hlin (always)
    float* bufB = take((size_t)GN * GD);      // agg ping
    float* bufC = take((size_t)GN * GD);      // agg pong
    float* z    = take(GB * ZW);              // [16,2176] concat feature
    float* z2   = take(GB * GD);              // [16,128] lin1 output

    const int nRowBlocks = GN / 32;           // 1500
    const size_t gemmLds =
        (size_t)(128 * WPAD + 2 * 32 * APAD) * sizeof(float); // 101,376 B

    // 1) init: zero degree + z accumulators (fresh every call)
    zero_f32<<<(GN + 255) / 256, 256, 0, stream>>>(dinv, GN);
    zero_f32<<<(GB * ZW + 255) / 256, 256, 0, stream>>>(z, GB * ZW);

    // 2) degrees -> dinv
    deg_count<<<(GE + 255) / 256, 256, 0, stream>>>(ei, dinv, GE);
    deg_rsqrt<<<(GN + 255) / 256, 256, 0, stream>>>(dinv, GN);

    // 3) five GCN layers: gemm -> self+bias init -> edge atomic scatter
    const float* hin = x;
    float* aggs[2] = { bufB, bufC };
    float* hcur = nullptr;
    for (int l = 0; l < 5; ++l) {
        float* agg = aggs[l & 1];
        gcn_gemm_wmma<<<600, 256, gemmLds, stream>>>(hin, Ws[l], bufA,
                                                     nRowBlocks, l > 0 ? 1 : 0);
        gcn_init_agg<<<(GN * 32 + 255) / 256, 256, 0, stream>>>(bufA, dinv, bs[l],
                                                                agg, GN * 32);
        gcn_edge_scatter<<<(GE * 32 + 255) / 256, 256, 0, stream>>>(bufA, dinv, ei,
                                                                    agg, GE);
        hin  = agg;
        hcur = agg;
    }

    // 4) pooling into z: base-node gather (cols 0..2047) + graph mean (cols 2048..2175)
    gather_base<<<GB * GK, GD, 0, stream>>>(hcur, base, z);
    pool_mean<<<dim3(GB, NPG / 125), GD, 0, stream>>>(hcur, z);

    // 5) heads
    lin1_wmma<<<1, 256, 0, stream>>>(z, lin1W, lin1b, z2);
    lin2_small<<<1, 64, 0, stream>>>(z2, cond, lin2W, lin2b, out);
}